// FieldAttentionalRNNDecoder_72782515798047
// MI455X (gfx1250) — compile-verified
//
#include <hip/hip_runtime.h>
#include <hip/hip_bf16.h>
#include <math.h>

// Problem constants (match reference)
#define B_   32
#define L_   64
#define LS_  100
#define H_   512
#define E_   512
#define F_   64
#define V_   32000
#define SOS_ 2
#define EPS_ 1e-6f

typedef __attribute__((ext_vector_type(2))) float v2f;
typedef __attribute__((ext_vector_type(8))) float v8f;

__device__ __forceinline__ float sigmf_(float x) { return 1.0f / (1.0f + expf(-x)); }

// ---------------------------------------------------------------------------
// Dual-M-tile fp32 WMMA accumulate:
//   C0 += A[0:16)  x B,   C1 += A[16:32) x B
// A: (32 x K) row-major (lda may be 0 to broadcast one row), B: (K x 16) row-major.
// Fragment layout per ISA (32b A 16x4 / B 4x16):
//   lane L: M(or N) = L%16 ; VGPR j holds K = 2*(L/16)+j
// B fragment is shared by both WMMAs -> halves B traffic per WMMA.
// K must be a multiple of 64 (all call sites use 512 or 64).
// ---------------------------------------------------------------------------
template <int K>
__device__ __forceinline__ void wmma_tile_2m(const float* __restrict__ A, int lda,
                                             const float* __restrict__ Bm, int ldb,
                                             v8f& acc0, v8f& acc1) {
  const int lane = threadIdx.x & 31;
  const int r16  = lane & 15;          // M row (A) / N col (B) within tile
  const int kg   = (lane >> 4) << 1;   // 0 or 2
  const float* ap0 = A + (size_t)r16 * (size_t)lda;
  const float* ap1 = A + (size_t)(r16 + 16) * (size_t)lda;
  for (int k0 = 0; k0 < K; k0 += 64) {
    if (k0 + 64 < K) {
      // stream-ahead hint on the weight panel (emits global_prefetch_b8)
      __builtin_prefetch(&Bm[(size_t)(k0 + 64 + lane) * (size_t)ldb + r16], 0, 1);
    }
#pragma unroll 8
    for (int kk = 0; kk < 64; kk += 4) {
      const int k = k0 + kk;
      v2f a0, a1, b;
      a0[0] = ap0[k + kg];
      a0[1] = ap0[k + kg + 1];
      a1[0] = ap1[k + kg];
      a1[1] = ap1[k + kg + 1];
      b[0]  = Bm[(size_t)(k + kg)     * (size_t)ldb + r16];
      b[1]  = Bm[(size_t)(k + kg + 1) * (size_t)ldb + r16];
      acc0 = __builtin_amdgcn_wmma_f32_16x16x4_f32(false, a0, false, b, (short)0, acc0,
                                                   false, false);
      acc1 = __builtin_amdgcn_wmma_f32_16x16x4_f32(false, a1, false, b, (short)0, acc1,
                                                   false, false);
    }
  }
}

// ---------------------------------------------------------------------------
// phi = tanh(A @ W + bias); A: (3200 x K) row-major, W: (K x 512), out: (3200 x 512)
// grid = (512/16, 3200/32), block = 32 (one wave -> 32x16 output)
// ---------------------------------------------------------------------------
template <int K>
__global__ void phi_kernel(const float* __restrict__ A,
                           const float* __restrict__ W, const float* __restrict__ bias,
                           float* __restrict__ out) {
  const int n0 = blockIdx.x * 16;
  const int m0 = blockIdx.y * 32;
  v8f acc0{}, acc1{};
  wmma_tile_2m<K>(A + (size_t)m0 * (size_t)K, K, W + n0, H_, acc0, acc1);
  const int lane = threadIdx.x & 31;
  const int n = n0 + (lane & 15);
  const int half = lane >> 4;
  const float bn = bias[n];
#pragma unroll
  for (int r = 0; r < 8; ++r) {
    const int m = m0 + half * 8 + r;
    out[(size_t)m * H_ + n]        = tanhf(acc0[r] + bn);
    out[(size_t)(m + 16) * H_ + n] = tanhf(acc1[r] + bn);
  }
}

// ---------------------------------------------------------------------------
// z = [x ; h_prev] @ lstm_W + lstm_b;  M=32, K=1024, N=2048
// x[m][k] = xptr[m*xstride + k]  (xstride==0 broadcasts embedding[SOS])
// grid = (2048/16), block = 32
// ---------------------------------------------------------------------------
__global__ void lstm_z_kernel(const float* __restrict__ xptr, int xstride,
                              const float* __restrict__ hbuf,
                              const float* __restrict__ W,
                              const float* __restrict__ bias,
                              float* __restrict__ z) {
  const int n0 = blockIdx.x * 16;
  v8f acc0{}, acc1{};
  wmma_tile_2m<E_>(xptr, xstride, W + n0, 4 * H_, acc0, acc1);
  wmma_tile_2m<H_>(hbuf, H_, W + (size_t)E_ * (4 * H_) + n0, 4 * H_, acc0, acc1);
  const int lane = threadIdx.x & 31;
  const int n = n0 + (lane & 15);
  const int half = lane >> 4;
  const float bn = bias[n];
#pragma unroll
  for (int r = 0; r < 8; ++r) {
    const int m = half * 8 + r;
    z[(size_t)m * (4 * H_) + n]        = acc0[r] + bn;
    z[(size_t)(m + 16) * (4 * H_) + n] = acc1[r] + bn;
  }
}

// ---------------------------------------------------------------------------
// LSTM gates + masking; elementwise over (B, H)
// ---------------------------------------------------------------------------
__global__ void gate_kernel(const float* __restrict__ z,
                            float* __restrict__ hbuf, float* __restrict__ cbuf,
                            float* __restrict__ otbuf,
                            const int* __restrict__ len, int t) {
  const int idx = blockIdx.x * blockDim.x + threadIdx.x;
  if (idx >= B_ * H_) return;
  const int m = idx / H_;
  const int n = idx % H_;
  const size_t zr = (size_t)m * (4 * H_);
  const float iv = z[zr + n];
  const float jv = z[zr + H_ + n];
  const float fv = z[zr + 2 * H_ + n];
  const float ov = z[zr + 3 * H_ + n];
  const float cp = cbuf[idx];
  const float hp = hbuf[idx];
  const float c = sigmf_(fv + 1.0f) * cp + sigmf_(iv) * tanhf(jv);
  const float h = sigmf_(ov) * tanhf(c);
  const bool fin = (t > 0) && ((t - 1) >= len[m]);
  otbuf[idx] = fin ? 0.0f : h;
  hbuf[idx]  = fin ? hp : h;
  cbuf[idx]  = fin ? cp : c;
}

// ---------------------------------------------------------------------------
// gamma = tanh(o_t @ Ws + bs), alpha = tanh(o_t @ Wr + br); selected by blockIdx.y
// grid = (512/16, 2), block = 32
// ---------------------------------------------------------------------------
__global__ void ga_kernel(const float* __restrict__ ot,
                          const float* __restrict__ Ws, const float* __restrict__ bs,
                          const float* __restrict__ Wr, const float* __restrict__ br,
                          float* __restrict__ gbuf, float* __restrict__ abuf) {
  const int n0 = blockIdx.x * 16;
  const float* W  = blockIdx.y ? Wr : Ws;
  const float* bi = blockIdx.y ? br : bs;
  float* out      = blockIdx.y ? abuf : gbuf;
  v8f acc0{}, acc1{};
  wmma_tile_2m<H_>(ot, H_, W + n0, H_, acc0, acc1);
  const int lane = threadIdx.x & 31;
  const int n = n0 + (lane & 15);
  const int half = lane >> 4;
  const float bn = bi[n];
#pragma unroll
  for (int r = 0; r < 8; ++r) {
    const int m = half * 8 + r;
    out[(size_t)m * H_ + n]        = tanhf(acc0[r] + bn);
    out[(size_t)(m + 16) * H_ + n] = tanhf(acc1[r] + bn);
  }
}

// ---------------------------------------------------------------------------
// Dual masked softmax attention + context; one block per batch row
// ---------------------------------------------------------------------------
__global__ void attention_kernel(const float* __restrict__ phi_hs,
                                 const float* __restrict__ phi_fds,
                                 const float* __restrict__ gbuf,
                                 const float* __restrict__ abuf,
                                 const float* __restrict__ enc,
                                 float* __restrict__ ctx) {
  const int b = blockIdx.x;
  __shared__ float sh[LS_];
  __shared__ float sf[LS_];
  __shared__ float w[LS_];
  const int tid = threadIdx.x;
  const float* g = gbuf + (size_t)b * H_;
  const float* a = abuf + (size_t)b * H_;
  for (int l = tid; l < LS_; l += blockDim.x) {
    const float* ph = phi_hs  + ((size_t)b * LS_ + l) * H_;
    const float* pf = phi_fds + ((size_t)b * LS_ + l) * H_;
    float acch = 0.0f, accf = 0.0f;
    for (int k = 0; k < H_; ++k) {
      acch += ph[k] * g[k];
      accf += pf[k] * a[k];
    }
    sh[l] = acch;
    sf[l] = accf;
  }
  __syncthreads();
  if (tid == 0) {
    float mh = -3.4e38f, mf = -3.4e38f;
    for (int l = 0; l < LS_; ++l) {
      mh = fmaxf(mh, sh[l]);
      mf = fmaxf(mf, sf[l]);
    }
    float Sh = 0.0f, Sf = 0.0f;
    for (int l = 0; l < LS_; ++l) {
      sh[l] = expf(sh[l] - mh); Sh += sh[l];
      sf[l] = expf(sf[l] - mf); Sf += sf[l];
    }
    const float ih  = 1.0f / (EPS_ + Sh);
    const float ifd = 1.0f / (EPS_ + Sf);
    float wsum = 0.0f;
    for (int l = 0; l < LS_; ++l) {
      w[l] = (sh[l] * ih) * (sf[l] * ifd);
      wsum += w[l];
    }
    const float inv = 1.0f / (EPS_ + wsum);
    for (int l = 0; l < LS_; ++l) w[l] *= inv;
  }
  __syncthreads();
  for (int n = tid; n < H_; n += blockDim.x) {
    float s = 0.0f;
    for (int l = 0; l < LS_; ++l)
      s += w[l] * enc[((size_t)b * LS_ + l) * H_ + n];
    ctx[(size_t)b * H_ + n] = s;
  }
}

// ---------------------------------------------------------------------------
// att = tanh([ctx ; o_t] @ Wo + bo), masked; M=32, K=1024, N=512
// grid = (512/16), block = 32
// ---------------------------------------------------------------------------
__global__ void att_kernel(const float* __restrict__ ctx, const float* __restrict__ ot,
                           const float* __restrict__ Wo, const float* __restrict__ bo,
                           float* __restrict__ att, const int* __restrict__ len, int t) {
  const int n0 = blockIdx.x * 16;
  v8f acc0{}, acc1{};
  wmma_tile_2m<H_>(ctx, H_, Wo + n0, H_, acc0, acc1);
  wmma_tile_2m<H_>(ot,  H_, Wo + (size_t)H_ * H_ + n0, H_, acc0, acc1);
  const int lane = threadIdx.x & 31;
  const int n = n0 + (lane & 15);
  const int half = lane >> 4;
  const float bn = bo[n];
#pragma unroll
  for (int r = 0; r < 8; ++r) {
    const int m  = half * 8 + r;
    const int m2 = m + 16;
    const bool fin  = (t > 0) && ((t - 1) >= len[m]);
    const bool fin2 = (t > 0) && ((t - 1) >= len[m2]);
    att[(size_t)m  * H_ + n] = fin  ? 0.0f : tanhf(acc0[r] + bn);
    att[(size_t)m2 * H_ + n] = fin2 ? 0.0f : tanhf(acc1[r] + bn);
  }
}

// ---------------------------------------------------------------------------
// logits = att @ out_W + out_b, masked, written to emit[b][t][:]
// M=32, K=512, N=32000 ; grid = (32000/16), block = 32
// ---------------------------------------------------------------------------
__global__ void logits_kernel(const float* __restrict__ att,
                              const float* __restrict__ outW,
                              const float* __restrict__ outb,
                              float* __restrict__ emit,
                              const int* __restrict__ len, int t) {
  const int n0 = blockIdx.x * 16;
  v8f acc0{}, acc1{};
  wmma_tile_2m<H_>(att, H_, outW + n0, V_, acc0, acc1);
  const int lane = threadIdx.x & 31;
  const int n = n0 + (lane & 15);
  const int half = lane >> 4;
  const float bn = outb[n];
#pragma unroll
  for (int r = 0; r < 8; ++r) {
    const int m  = half * 8 + r;
    const int m2 = m + 16;
    const bool fin  = (t > 0) && ((t - 1) >= len[m]);
    const bool fin2 = (t > 0) && ((t - 1) >= len[m2]);
    emit[(size_t)m  * ((L_ + 1) * (size_t)V_) + (size_t)t * V_ + n] =
        fin  ? 0.0f : (acc0[r] + bn);
    emit[(size_t)m2 * ((L_ + 1) * (size_t)V_) + (size_t)t * V_ + n] =
        fin2 ? 0.0f : (acc1[r] + bn);
  }
}

// ---------------------------------------------------------------------------
// Final carry copy: [h ; c] -> tail of d_out
// ---------------------------------------------------------------------------
__global__ void copy_hc_kernel(const float* __restrict__ hbuf,
                               const float* __restrict__ cbuf,
                               float* __restrict__ dst) {
  const int idx = blockIdx.x * blockDim.x + threadIdx.x;
  if (idx < B_ * H_) dst[idx] = hbuf[idx];
  else if (idx < 2 * B_ * H_) dst[idx] = cbuf[idx - B_ * H_];
}

extern "C" void kernel_launch(void* const* d_in, const int* in_sizes, int n_in,
                              void* d_out, int out_size, void* d_ws, size_t ws_size,
                              hipStream_t stream) {
  const float* h0    = (const float*)d_in[0];
  const float* c0    = (const float*)d_in[1];
  const float* ie    = (const float*)d_in[2];   // (B, L, E)
  const float* enc   = (const float*)d_in[3];   // (B, LS, H)
  const float* fld   = (const float*)d_in[4];   // (B, LS, F)
  const float* emb   = (const float*)d_in[5];   // (V, E)
  const float* lstmW = (const float*)d_in[6];   // (E+H, 4H)
  const float* lstmB = (const float*)d_in[7];
  const float* Wh    = (const float*)d_in[8];
  const float* bh    = (const float*)d_in[9];
  const float* Ws    = (const float*)d_in[10];
  const float* bs    = (const float*)d_in[11];
  const float* Wr    = (const float*)d_in[12];
  const float* br    = (const float*)d_in[13];
  const float* Wf    = (const float*)d_in[14];
  const float* bf    = (const float*)d_in[15];
  const float* Wo    = (const float*)d_in[16];  // (2H, H)
  const float* bo    = (const float*)d_in[17];
  const float* outW  = (const float*)d_in[18];  // (H, V)
  const float* outb  = (const float*)d_in[19];
  const int*   len   = (const int*)d_in[20];

  float* emit = (float*)d_out;                  // (B, L+1, V) + h + c
  float* ws   = (float*)d_ws;                   // needs ~13.9 MB

  float* phi_hs  = ws;
  float* phi_fds = phi_hs  + (size_t)B_ * LS_ * H_;
  float* hbuf    = phi_fds + (size_t)B_ * LS_ * H_;
  float* cbuf    = hbuf  + (size_t)B_ * H_;
  float* otbuf   = cbuf  + (size_t)B_ * H_;
  float* gbuf    = otbuf + (size_t)B_ * H_;
  float* abuf    = gbuf  + (size_t)B_ * H_;
  float* zbuf    = abuf  + (size_t)B_ * H_;     // (B, 4H)
  float* ctx     = zbuf  + (size_t)B_ * 4 * H_;
  float* att     = ctx   + (size_t)B_ * H_;

  // carry init (graph-capture-safe d2d async copies)
  hipMemcpyAsync(hbuf, h0, (size_t)B_ * H_ * sizeof(float), hipMemcpyDeviceToDevice, stream);
  hipMemcpyAsync(cbuf, c0, (size_t)B_ * H_ * sizeof(float), hipMemcpyDeviceToDevice, stream);

  // one-time precompute of phi tensors
  phi_kernel<H_><<<dim3(H_ / 16, (B_ * LS_) / 32), 32, 0, stream>>>(enc, Wh, bh, phi_hs);
  phi_kernel<F_><<<dim3(H_ / 16, (B_ * LS_) / 32), 32, 0, stream>>>(fld, Wf, bf, phi_fds);

  for (int t = 0; t <= L_; ++t) {
    const float* xptr;
    int xstride;
    if (t == 0) { xptr = emb + (size_t)SOS_ * E_; xstride = 0; }
    else        { xptr = ie + (size_t)(t - 1) * E_; xstride = L_ * E_; }

    lstm_z_kernel<<<4 * H_ / 16, 32, 0, stream>>>(xptr, xstride, hbuf, lstmW, lstmB, zbuf);
    gate_kernel<<<(B_ * H_ + 255) / 256, 256, 0, stream>>>(zbuf, hbuf, cbuf, otbuf, len, t);
    ga_kernel<<<dim3(H_ / 16, 2), 32, 0, stream>>>(otbuf, Ws, bs, Wr, br, gbuf, abuf);
    attention_kernel<<<B_, 128, 0, stream>>>(phi_hs, phi_fds, gbuf, abuf, enc, ctx);
    att_kernel<<<H_ / 16, 32, 0, stream>>>(ctx, otbuf, Wo, bo, att, len, t);
    logits_kernel<<<V_ / 16, 32, 0, stream>>>(att, outW, outb, emit, len, t);
  }

  copy_hc_kernel<<<(2 * B_ * H_ + 255) / 256, 256, 0, stream>>>(
      hbuf, cbuf, emit + (size_t)B_ * (L_ + 1) * V_);
}